// CombinedKVCacheCompression_88562225643885
// MI455X (gfx1250) — compile-verified
//
#include <hip/hip_runtime.h>
#include <stdint.h>

#define BLOCK   256
#define PTV     4                         // float4 per thread per tile
#define TILE_V4 (BLOCK * PTV)             // 1024 float4 = 4096 floats per tile
#define MAXNB   1024                      // max blocks (tiles grouped contiguously)

// ws layout (uint32 units):
//  [0]=min(enc) [1]=max(enc) [2]=t [3]=rem
//  [4..259]            global 256-bin histogram
//  [260..260+nb)       per-block exclusive prefix of eq-counts
//  [260+nb..260+2nb)   per-block eq-counts (fallback path)
//  [260+2nb.. )        per-block 256-bin histograms (fast path, nb*256)
#define WS_MIN  0
#define WS_MAX  1
#define WS_T    2
#define WS_REM  3
#define WS_HIST 4
#define WS_PRE  260

// ---- order-preserving float<->uint encoding (for integer atomic min/max) ----
__device__ __forceinline__ unsigned encf(float f) {
    unsigned b = __float_as_uint(f);
    return (b & 0x80000000u) ? ~b : (b | 0x80000000u);
}
__device__ __forceinline__ float decf(unsigned m) {
    unsigned b = (m & 0x80000000u) ? (m ^ 0x80000000u) : ~m;
    return __uint_as_float(b);
}
__device__ __forceinline__ float quantf(float x, float mn, float s) {
    return rintf((x - mn) * s);           // RNE, matches jnp.round
}
__device__ __forceinline__ unsigned quantu(float qf) {
    int q = (int)qf;
    q = q < 0 ? 0 : (q > 255 ? 255 : q);
    return (unsigned)q;
}
// contiguous balanced tile range for block b (identical in every kernel)
__device__ __forceinline__ void tile_range(int b, int tiles, int nb, int& t0, int& t1) {
    t0 = (int)(((long long)b) * tiles / nb);
    t1 = (int)(((long long)(b + 1)) * tiles / nb);
}

// ---------------- K0: init workspace ----------------
__global__ void k0_init(unsigned* ws) {
    int t = threadIdx.x;
    if (t == 0) { ws[WS_MIN] = 0xFFFFFFFFu; ws[WS_MAX] = 0u; }
    ws[WS_HIST + t] = 0u;
}

// ---------------- K1: min/max, double-buffered async-to-LDS staging ----------------
__global__ void k1_minmax(const float4* __restrict__ in, unsigned* ws,
                          int nvec4, int tiles, int nb) {
    __shared__ float4 stage[2][TILE_V4];              // 32 KiB, LDS offset 0
    int tid = threadIdx.x;
    int t0, t1; tile_range(blockIdx.x, tiles, nb, t0, t1);

    float vmin = 3.402823466e38f, vmax = -3.402823466e38f;

    // prologue: stage first tile if it is fully in range (uniform predicate)
    bool f0 = (t0 < t1) && ((long long)(t0 + 1) * TILE_V4 <= nvec4);
    if (f0) {
#pragma unroll
        for (int i = 0; i < PTV; ++i) {
            long long v4 = (long long)t0 * TILE_V4 + i * BLOCK + tid;
            unsigned ldsoff = (unsigned)((i * BLOCK + tid) * sizeof(float4));
            const float4* g = in + v4;
            asm volatile("global_load_async_to_lds_b128 %0, %1, off"
                         :: "v"(ldsoff), "v"(g) : "memory");
        }
    }
    for (int t = t0; t < t1; ++t) {
        int cur = (t - t0) & 1;
        bool fcur  = ((long long)(t + 1) * TILE_V4 <= nvec4);
        bool fnext = (t + 1 < t1) && ((long long)(t + 2) * TILE_V4 <= nvec4);
        if (fnext) {                                   // issue next tile into other buffer
            unsigned bufbase = (unsigned)((1 - cur) * TILE_V4 * sizeof(float4));
#pragma unroll
            for (int i = 0; i < PTV; ++i) {
                long long v4 = (long long)(t + 1) * TILE_V4 + i * BLOCK + tid;
                unsigned ldsoff = bufbase + (unsigned)((i * BLOCK + tid) * sizeof(float4));
                const float4* g = in + v4;
                asm volatile("global_load_async_to_lds_b128 %0, %1, off"
                             :: "v"(ldsoff), "v"(g) : "memory");
            }
        }
        if (fcur) {
            if (fnext) asm volatile("s_wait_asynccnt 4" ::: "memory");   // oldest 4 done
            else       asm volatile("s_wait_asynccnt 0" ::: "memory");
            __syncthreads();
#pragma unroll
            for (int i = 0; i < PTV; ++i) {
                float4 d = stage[cur][i * BLOCK + tid];
                vmin = fminf(vmin, fminf(fminf(d.x, d.y), fminf(d.z, d.w)));
                vmax = fmaxf(vmax, fmaxf(fmaxf(d.x, d.y), fmaxf(d.z, d.w)));
            }
        } else {                                       // partial tail tile: direct loads
#pragma unroll
            for (int i = 0; i < PTV; ++i) {
                long long v4 = (long long)t * TILE_V4 + i * BLOCK + tid;
                if (v4 < nvec4) {
                    float4 d = in[v4];
                    vmin = fminf(vmin, fminf(fminf(d.x, d.y), fminf(d.z, d.w)));
                    vmax = fmaxf(vmax, fmaxf(fmaxf(d.x, d.y), fmaxf(d.z, d.w)));
                }
            }
        }
        __syncthreads();                               // buf reuse two tiles later
    }
    // wave32 reduction, then cross-wave via LDS (reuse stage as scratch)
    for (int o = 16; o > 0; o >>= 1) {
        vmin = fminf(vmin, __shfl_xor(vmin, o, 32));
        vmax = fmaxf(vmax, __shfl_xor(vmax, o, 32));
    }
    int lane = tid & 31, wid = tid >> 5;
    unsigned* red = (unsigned*)stage;
    if (lane == 0) { red[wid] = encf(vmin); red[8 + wid] = encf(vmax); }
    __syncthreads();
    if (tid == 0) {
        unsigned mn = red[0], mx = red[8];
        for (int w = 1; w < 8; ++w) { mn = min(mn, red[w]); mx = max(mx, red[8 + w]); }
        atomicMin(&ws[WS_MIN], mn);
        atomicMax(&ws[WS_MAX], mx);
    }
}

// ---------------- K2: histogram (global + optional per-block store) ----------------
__global__ void k2_hist(const float4* __restrict__ in, unsigned* ws, unsigned* bh,
                        int nvec4, int tiles, int nb) {
    __shared__ unsigned h[8][256];                    // per-wave sub-histograms
    int tid = threadIdx.x, wid = tid >> 5;
    for (int w = 0; w < 8; ++w) h[w][tid] = 0;
    __syncthreads();

    float mn = decf(ws[WS_MIN]), mx = decf(ws[WS_MAX]);
    float rng = mx - mn; if (rng == 0.0f) rng = 1.0f;
    float s = 255.0f / rng;

    int t0, t1; tile_range(blockIdx.x, tiles, nb, t0, t1);
    for (int t = t0; t < t1; ++t) {
#pragma unroll
        for (int i = 0; i < PTV; ++i) {
            long long v4 = (long long)t * TILE_V4 + i * BLOCK + tid;
            if (v4 < nvec4) {
                float4 d = in[v4];
                atomicAdd(&h[wid][quantu(quantf(d.x, mn, s))], 1u);
                atomicAdd(&h[wid][quantu(quantf(d.y, mn, s))], 1u);
                atomicAdd(&h[wid][quantu(quantf(d.z, mn, s))], 1u);
                atomicAdd(&h[wid][quantu(quantf(d.w, mn, s))], 1u);
            }
        }
    }
    __syncthreads();
    unsigned c = 0;
    for (int w = 0; w < 8; ++w) c += h[w][tid];
    if (bh) bh[(size_t)blockIdx.x * 256 + tid] = c;   // fast path: per-block histogram
    if (c) atomicAdd(&ws[WS_HIST + tid], c);
}

// -------- threshold helper (one thread) --------
__device__ __forceinline__ void do_threshold(unsigned* ws, float* out_tail, unsigned keep) {
    unsigned cum = 0, t = 0, rem = keep;
    for (int v = 255; v >= 0; --v) {
        unsigned hv = ws[WS_HIST + v];
        cum += hv;
        if (cum >= keep) { t = (unsigned)v; rem = keep - (cum - hv); break; }
    }
    ws[WS_T] = t; ws[WS_REM] = rem;
    float mn = decf(ws[WS_MIN]), mx = decf(ws[WS_MAX]);
    float rng = mx - mn; if (rng == 0.0f) rng = 1.0f;
    out_tail[0] = mn; out_tail[1] = mx; out_tail[2] = rng;
}

// ---------------- K5big: threshold + scan of per-block hist column t ----------------
__global__ void k5_big(unsigned* ws, const unsigned* __restrict__ bh,
                       float* out_tail, unsigned keep, int nb) {
    __shared__ unsigned tsum[256];
    __shared__ unsigned tsh;
    int tid = threadIdx.x;
    if (tid == 0) { do_threshold(ws, out_tail, keep); tsh = ws[WS_T]; }
    __syncthreads();
    unsigned t = tsh;
    int per = (nb + 255) / 256;
    int lo = tid * per, hi = min(lo + per, nb);
    unsigned s = 0;
    for (int b = lo; b < hi; ++b) s += bh[(size_t)b * 256 + t];
    tsum[tid] = s;
    __syncthreads();
    unsigned run = 0;
    for (int i = 0; i < tid; ++i) run += tsum[i];
    for (int b = lo; b < hi; ++b) { ws[WS_PRE + b] = run; run += bh[(size_t)b * 256 + t]; }
}

// ---------------- fallback path: K3 threshold, K4 eq-count, K5small scan ----------------
__global__ void k3_thresh(unsigned* ws, float* out_tail, unsigned keep) {
    if (threadIdx.x == 0 && blockIdx.x == 0) do_threshold(ws, out_tail, keep);
}

__global__ void k4_eqcount(const float4* __restrict__ in, unsigned* ws,
                           int nvec4, int tiles, int nb) {
    __shared__ unsigned red[8];
    int tid = threadIdx.x, lane = tid & 31, wid = tid >> 5;
    float mn = decf(ws[WS_MIN]), mx = decf(ws[WS_MAX]);
    float rng = mx - mn; if (rng == 0.0f) rng = 1.0f;
    float s = 255.0f / rng;
    unsigned t = ws[WS_T];

    unsigned c = 0;
    int t0, t1; tile_range(blockIdx.x, tiles, nb, t0, t1);
    for (int tt = t0; tt < t1; ++tt) {
#pragma unroll
        for (int i = 0; i < PTV; ++i) {
            long long v4 = (long long)tt * TILE_V4 + i * BLOCK + tid;
            if (v4 < nvec4) {
                float4 d = in[v4];
                c += (quantu(quantf(d.x, mn, s)) == t);
                c += (quantu(quantf(d.y, mn, s)) == t);
                c += (quantu(quantf(d.z, mn, s)) == t);
                c += (quantu(quantf(d.w, mn, s)) == t);
            }
        }
    }
    for (int o = 16; o > 0; o >>= 1) c += __shfl_xor(c, o, 32);
    if (lane == 0) red[wid] = c;
    __syncthreads();
    if (tid == 0) {
        unsigned sum = 0;
        for (int w = 0; w < 8; ++w) sum += red[w];
        ws[WS_PRE + nb + blockIdx.x] = sum;           // eq-count slot
    }
}

__global__ void k5_small(unsigned* ws, int nb) {
    __shared__ unsigned tsum[256];
    int tid = threadIdx.x;
    const unsigned* eq = ws + WS_PRE + nb;
    int per = (nb + 255) / 256;
    int lo = tid * per, hi = min(lo + per, nb);
    unsigned s = 0;
    for (int b = lo; b < hi; ++b) s += eq[b];
    tsum[tid] = s;
    __syncthreads();
    unsigned run = 0;
    for (int i = 0; i < tid; ++i) run += tsum[i];
    for (int b = lo; b < hi; ++b) { ws[WS_PRE + b] = run; run += eq[b]; }
}

// ---------------- K6: final sparsified write with ordered tie-break ----------------
__global__ void k6_write(const float4* __restrict__ in, float4* __restrict__ out,
                         const unsigned* __restrict__ ws, int nvec4, int tiles, int nb) {
    __shared__ unsigned wsum[8];
    int tid = threadIdx.x, lane = tid & 31, wid = tid >> 5;
    float mn = decf(ws[WS_MIN]), mx = decf(ws[WS_MAX]);
    float rng = mx - mn; if (rng == 0.0f) rng = 1.0f;
    float s = 255.0f / rng;
    unsigned t = ws[WS_T], rem = ws[WS_REM];
    unsigned blockpre = ws[WS_PRE + blockIdx.x];
    unsigned running = 0;                              // identical in every thread

    int t0, t1; tile_range(blockIdx.x, tiles, nb, t0, t1);
    for (int tt = t0; tt < t1; ++tt) {
#pragma unroll
        for (int i = 0; i < PTV; ++i) {
            long long v4 = (long long)tt * TILE_V4 + i * BLOCK + tid;
            bool valid = v4 < nvec4;
            float4 d = valid ? in[v4] : make_float4(0.f, 0.f, 0.f, 0.f);
            float q0 = quantf(d.x, mn, s), q1 = quantf(d.y, mn, s);
            float q2 = quantf(d.z, mn, s), q3 = quantf(d.w, mn, s);
            unsigned u0 = quantu(q0), u1 = quantu(q1), u2 = quantu(q2), u3 = quantu(q3);
            unsigned e0 = (valid && u0 == t), e1 = (valid && u1 == t);
            unsigned e2 = (valid && u2 == t), e3 = (valid && u3 == t);
            unsigned c = e0 + e1 + e2 + e3;

            unsigned inc = c;                          // wave32 inclusive scan
            for (int o = 1; o < 32; o <<= 1) {
                unsigned n = __shfl_up(inc, o, 32);
                if (lane >= o) inc += n;
            }
            unsigned excl = inc - c;
            if (lane == 31) wsum[wid] = inc;
            __syncthreads();

            unsigned woff = 0, tot = 0;
            for (int w = 0; w < 8; ++w) { if (w < wid) woff += wsum[w]; tot += wsum[w]; }
            unsigned r = blockpre + running + woff + excl;

            float4 o4;
            o4.x = (valid && (u0 > t || (e0 && r < rem))) ? q0 : 0.f; r += e0;
            o4.y = (valid && (u1 > t || (e1 && r < rem))) ? q1 : 0.f; r += e1;
            o4.z = (valid && (u2 > t || (e2 && r < rem))) ? q2 : 0.f; r += e2;
            o4.w = (valid && (u3 > t || (e3 && r < rem))) ? q3 : 0.f;
            if (valid) out[v4] = o4;

            running += tot;
            __syncthreads();                           // wsum reused next iteration
        }
    }
}

extern "C" void kernel_launch(void* const* d_in, const int* in_sizes, int n_in,
                              void* d_out, int out_size, void* d_ws, size_t ws_size,
                              hipStream_t stream) {
    const float* x = (const float*)d_in[0];
    int N = in_sizes[0];
    float* out = (float*)d_out;
    unsigned* ws = (unsigned*)d_ws;

    int nvec4 = N / 4;                                 // N = 16,777,216
    int tiles = (nvec4 + TILE_V4 - 1) / TILE_V4;       // 4096 tiles
    int nb = tiles < MAXNB ? tiles : MAXNB;            // 1024 blocks, 4 tiles each
    unsigned keep = (unsigned)(N / 2);                 // int(total * (1 - 0.5))

    size_t need_small = (size_t)(WS_PRE + 2 * nb) * sizeof(unsigned);
    size_t need_big   = need_small + (size_t)nb * 256 * sizeof(unsigned);
    bool big = ws_size >= need_big;
    unsigned* bh = big ? (ws + WS_PRE + 2 * nb) : nullptr;

    k0_init   <<<1, 256, 0, stream>>>(ws);
    k1_minmax <<<nb, BLOCK, 0, stream>>>((const float4*)x, ws, nvec4, tiles, nb);
    k2_hist   <<<nb, BLOCK, 0, stream>>>((const float4*)x, ws, bh, nvec4, tiles, nb);
    if (big) {
        k5_big  <<<1, 256, 0, stream>>>(ws, bh, out + N, keep, nb);
    } else {
        k3_thresh <<<1, 32, 0, stream>>>(ws, out + N, keep);
        k4_eqcount<<<nb, BLOCK, 0, stream>>>((const float4*)x, ws, nvec4, tiles, nb);
        k5_small  <<<1, 256, 0, stream>>>(ws, nb);
    }
    k6_write  <<<nb, BLOCK, 0, stream>>>((const float4*)x, (float4*)out, ws,
                                         nvec4, tiles, nb);
}